// GraphSAGEEncoder_53163105190283
// MI455X (gfx1250) — compile-verified
//
#include <hip/hip_runtime.h>

typedef __attribute__((ext_vector_type(2))) float v2f;
typedef __attribute__((ext_vector_type(8))) float v8f;

#define NODES 50000
#define EDGES 800000
#define DIN   128
#define DHID  128
#define DOUT  256

// ---------------- CSR construction ----------------

__global__ void deg_count_kernel(const int* __restrict__ dst, int* __restrict__ deg, int e) {
    int i = blockIdx.x * blockDim.x + threadIdx.x;
    if (i < e) atomicAdd(&deg[dst[i]], 1);
}

// Single-block exclusive scan (Hillis-Steele tiles + running carry).
__global__ void scan_kernel(const int* __restrict__ deg, int* __restrict__ offs, int n) {
    __shared__ int buf[1024];
    __shared__ int carry;
    if (threadIdx.x == 0) carry = 0;
    __syncthreads();
    for (int base = 0; base < n; base += 1024) {
        int i = base + (int)threadIdx.x;
        int v = (i < n) ? deg[i] : 0;
        buf[threadIdx.x] = v;
        __syncthreads();
        for (int ofs = 1; ofs < 1024; ofs <<= 1) {
            int t = (threadIdx.x >= (unsigned)ofs) ? buf[threadIdx.x - ofs] : 0;
            __syncthreads();
            buf[threadIdx.x] += t;
            __syncthreads();
        }
        int incl = buf[threadIdx.x];
        if (i < n) offs[i] = carry + incl - v;   // exclusive
        __syncthreads();
        if (threadIdx.x == 1023) carry += buf[1023];
        __syncthreads();
    }
}

__global__ void fill_kernel(const int* __restrict__ src, const int* __restrict__ dst,
                            int* __restrict__ cur, int* __restrict__ csr, int e) {
    int i = blockIdx.x * blockDim.x + threadIdx.x;
    if (i < e) {
        int p = atomicAdd(&cur[dst[i]], 1);
        csr[p] = src[i];
    }
}

// ---------------- mean aggregation: one wave per node ----------------
// 32 lanes x float4 = one full 512B feature row per edge, coalesced gather.

__global__ void agg_mean_kernel(const float* __restrict__ X, const int* __restrict__ offs,
                                const int* __restrict__ deg, const int* __restrict__ csr,
                                float* __restrict__ MEAN, int n) {
    int gid  = blockIdx.x * blockDim.x + threadIdx.x;
    int node = gid >> 5;
    int lane = gid & 31;
    if (node >= n) return;
    int start = offs[node];
    int d     = deg[node];
    int fo    = lane * 4;
    float4 acc = make_float4(0.f, 0.f, 0.f, 0.f);
    for (int j = 0; j < d; ++j) {
        int s = csr[start + j];
        float4 v = *reinterpret_cast<const float4*>(X + (size_t)s * DIN + fo);
        acc.x += v.x; acc.y += v.y; acc.z += v.z; acc.w += v.w;
    }
    float inv = (d > 0) ? (1.0f / (float)d) : 0.0f;   // mean / max(cnt,1); agg==0 when d==0
    acc.x *= inv; acc.y *= inv; acc.z *= inv; acc.w *= inv;
    *reinterpret_cast<float4*>(MEAN + (size_t)node * DIN + fo) = acc;
}

// ---------------- fused dual-GEMM + bias (+ReLU) via fp32 WMMA ----------------
// OUT[m, c] = act( sum_k MEAN[m,k]*Wl[c,k] + bl[c] + sum_k X[m,k]*Wr[c,k] )
// One wave per 16x16 output tile, V_WMMA_F32_16X16X4_F32, K step = 4.

template <bool RELU>
__global__ void sage_gemm_kernel(const float* __restrict__ X,
                                 const float* __restrict__ MEAN,
                                 const float* __restrict__ Wl,
                                 const float* __restrict__ bl,
                                 const float* __restrict__ Wr,
                                 float* __restrict__ OUT,
                                 int n, int dout) {
    const int lane = threadIdx.x & 31;
    const int wave = threadIdx.x >> 5;
    const int col_tiles   = dout >> 4;
    const int total_tiles = (n >> 4) * col_tiles;
    const int tile = blockIdx.x * (blockDim.x >> 5) + wave;
    if (tile >= total_tiles) return;              // wave-uniform: EXEC stays all-1s

    const int row_tile = tile / col_tiles;
    const int col_tile = tile - row_tile * col_tiles;
    const int m0 = row_tile << 4;
    const int n0 = col_tile << 4;

    const int l15  = lane & 15;
    const int koff = (lane >> 4) << 1;            // lanes 0-15 -> K 0,1 ; lanes 16-31 -> K 2,3
    const size_t arow = (size_t)(m0 + l15) * DIN; // A: M = lane%16
    const size_t brow = (size_t)(n0 + l15) * DIN; // B col N = lane%16, B[k][n] = W[n*DIN + k]

    v8f c = {0.f, 0.f, 0.f, 0.f, 0.f, 0.f, 0.f, 0.f};

    // pass 1: MEAN @ Wl^T
#pragma unroll 4
    for (int k0 = 0; k0 < DIN; k0 += 4) {
        float2 av = *reinterpret_cast<const float2*>(MEAN + arow + k0 + koff);
        float2 bv = *reinterpret_cast<const float2*>(Wl   + brow + k0 + koff);
        v2f a; a.x = av.x; a.y = av.y;
        v2f b; b.x = bv.x; b.y = bv.y;
        c = __builtin_amdgcn_wmma_f32_16x16x4_f32(false, a, false, b, (short)0, c, false, false);
    }
    // pass 2: X @ Wr^T (accumulates into same C)
#pragma unroll 4
    for (int k0 = 0; k0 < DIN; k0 += 4) {
        float2 av = *reinterpret_cast<const float2*>(X  + arow + k0 + koff);
        float2 bv = *reinterpret_cast<const float2*>(Wr + brow + k0 + koff);
        v2f a; a.x = av.x; a.y = av.y;
        v2f b; b.x = bv.x; b.y = bv.y;
        c = __builtin_amdgcn_wmma_f32_16x16x4_f32(false, a, false, b, (short)0, c, false, false);
    }

    // C/D layout: VGPR r -> row m0+r (lanes 0-15) / m0+r+8 (lanes 16-31), col n0 + lane%16
    const float bias    = bl[n0 + l15];
    const int   rowbase = m0 + ((lane >> 4) << 3);
    const int   col     = n0 + l15;
#pragma unroll
    for (int r = 0; r < 8; ++r) {
        float v = c[r] + bias;
        if (RELU) v = fmaxf(v, 0.0f);
        OUT[(size_t)(rowbase + r) * dout + col] = v;
    }
}

// ---------------- launch ----------------

extern "C" void kernel_launch(void* const* d_in, const int* in_sizes, int n_in,
                              void* d_out, int out_size, void* d_ws, size_t ws_size,
                              hipStream_t stream) {
    (void)in_sizes; (void)n_in; (void)out_size; (void)ws_size;

    const float* x   = (const float*)d_in[0];
    const int*   ei  = (const int*)  d_in[1];   // [2, E] int32
    const float* Wl0 = (const float*)d_in[2];
    const float* bl0 = (const float*)d_in[3];
    const float* Wr0 = (const float*)d_in[4];
    const float* Wl1 = (const float*)d_in[5];
    const float* bl1 = (const float*)d_in[6];
    const float* Wr1 = (const float*)d_in[7];
    const float* Wl2 = (const float*)d_in[8];
    const float* bl2 = (const float*)d_in[9];
    const float* Wr2 = (const float*)d_in[10];

    const int* src = ei;
    const int* dst = ei + EDGES;

    char* ws = (char*)d_ws;
    size_t off = 0;
    float* mean = (float*)(ws + off); off += (size_t)NODES * DIN * sizeof(float);
    float* h1   = (float*)(ws + off); off += (size_t)NODES * DIN * sizeof(float);
    float* h2   = (float*)(ws + off); off += (size_t)NODES * DIN * sizeof(float);
    int*   deg  = (int*)  (ws + off); off += (size_t)NODES * sizeof(int);
    int*   offs = (int*)  (ws + off); off += (size_t)NODES * sizeof(int);
    int*   cur  = (int*)  (ws + off); off += (size_t)NODES * sizeof(int);
    int*   csr  = (int*)  (ws + off); off += (size_t)EDGES * sizeof(int);

    // Build CSR (by destination) each call — deterministic, capture-safe.
    hipMemsetAsync(deg, 0, (size_t)NODES * sizeof(int), stream);
    deg_count_kernel<<<(EDGES + 255) / 256, 256, 0, stream>>>(dst, deg, EDGES);
    scan_kernel<<<1, 1024, 0, stream>>>(deg, offs, NODES);
    hipMemcpyAsync(cur, offs, (size_t)NODES * sizeof(int), hipMemcpyDeviceToDevice, stream);
    fill_kernel<<<(EDGES + 255) / 256, 256, 0, stream>>>(src, dst, cur, csr, EDGES);

    const int aggBlocks = (NODES * 32 + 255) / 256;

    // Layer 0: h1 = relu(mean(x) @ Wl0^T + bl0 + x @ Wr0^T)
    agg_mean_kernel<<<aggBlocks, 256, 0, stream>>>(x, offs, deg, csr, mean, NODES);
    {
        int tiles = (NODES / 16) * (DHID / 16);
        sage_gemm_kernel<true><<<(tiles + 3) / 4, 128, 0, stream>>>(x, mean, Wl0, bl0, Wr0, h1, NODES, DHID);
    }
    // Layer 1
    agg_mean_kernel<<<aggBlocks, 256, 0, stream>>>(h1, offs, deg, csr, mean, NODES);
    {
        int tiles = (NODES / 16) * (DHID / 16);
        sage_gemm_kernel<true><<<(tiles + 3) / 4, 128, 0, stream>>>(h1, mean, Wl1, bl1, Wr1, h2, NODES, DHID);
    }
    // Layer 2 (no ReLU, DOUT=256) -> d_out
    agg_mean_kernel<<<aggBlocks, 256, 0, stream>>>(h2, offs, deg, csr, mean, NODES);
    {
        int tiles = (NODES / 16) * (DOUT / 16);
        sage_gemm_kernel<false><<<(tiles + 3) / 4, 128, 0, stream>>>(h2, mean, Wl2, bl2, Wr2, (float*)d_out, NODES, DOUT);
    }
}